// QuantumPINN_33036888441100
// MI455X (gfx1250) — compile-verified
//
#include <hip/hip_runtime.h>
#include <cstdint>

#define BLOCK 256
#define WPB (BLOCK / 32)          // waves per block (wave32)
#define GRID_BLOCKS 2048
#define DEPTH 4                   // async pipeline depth (buffers per wave)

typedef float v4f __attribute__((ext_vector_type(4)));

// ---------------- tiny complex 2x2 helpers (coefficient setup) ----------------
struct C2 { float re, im; };
__device__ __forceinline__ C2 cmul(C2 a, C2 b) {
    return { fmaf(a.re, b.re, -a.im * b.im), fmaf(a.re, b.im, a.im * b.re) };
}
__device__ __forceinline__ C2 cadd(C2 a, C2 b) { return { a.re + b.re, a.im + b.im }; }

__device__ __forceinline__ void mm2(C2 o[2][2], const C2 A[2][2], const C2 B[2][2]) {
    C2 t[2][2];
#pragma unroll
    for (int i = 0; i < 2; ++i)
#pragma unroll
        for (int j = 0; j < 2; ++j)
            t[i][j] = cadd(cmul(A[i][0], B[0][j]), cmul(A[i][1], B[1][j]));
#pragma unroll
    for (int i = 0; i < 2; ++i)
#pragma unroll
        for (int j = 0; j < 2; ++j)
            o[i][j] = t[i][j];
}

// ---------------- CDNA5 async global->LDS (ASYNCcnt-tracked) ----------------
__device__ __forceinline__ void async_ld_b128(unsigned lds_off, const v4f* gaddr) {
    asm volatile("global_load_async_to_lds_b128 %0, %1, off"
                 :: "v"(lds_off), "v"((unsigned long long)(uintptr_t)gaddr)
                 : "memory");
}
// wait until ASYNCcnt <= n (immediate must be a literal)
__device__ __forceinline__ void wait_async(int n) {
    if (n <= 0)      asm volatile("s_wait_asynccnt 0" ::: "memory");
    else if (n == 1) asm volatile("s_wait_asynccnt 1" ::: "memory");
    else if (n == 2) asm volatile("s_wait_asynccnt 2" ::: "memory");
    else             asm volatile("s_wait_asynccnt 3" ::: "memory");
}

__device__ __forceinline__ float eval1(float xv, float k0, float kc, float ks) {
    // out = k0 + kc*cos(x) + ks*sin(x); hardware trans ops v_cos_f32 / v_sin_f32
    return fmaf(kc, __cosf(xv), fmaf(ks, __sinf(xv), k0));
}

__global__ __launch_bounds__(BLOCK)
void qpinn_stream_kernel(const float* __restrict__ x,
                         const float* __restrict__ w, int L,
                         const float* __restrict__ pa, const float* __restrict__ pb,
                         float* __restrict__ out, int N) {
    // -------- wave-uniform coefficient computation (fast HW trig; tiny angles) --------
    C2 U[2][2] = { { {1.f, 0.f}, {0.f, 0.f} }, { {0.f, 0.f}, {1.f, 0.f} } };
    for (int l = 0; l < L; ++l) {
        float h0 = 0.5f * w[3 * l + 0];
        float h1 = 0.5f * w[3 * l + 1];
        float h2 = 0.5f * w[3 * l + 2];
        float c0 = __cosf(h0), s0 = __sinf(h0);
        float c1 = __cosf(h1), s1 = __sinf(h1);
        float c2 = __cosf(h2), s2 = __sinf(h2);
        C2 RX[2][2] = { { {c0, 0.f}, {0.f, -s0} }, { {0.f, -s0}, {c0, 0.f} } };
        C2 RY[2][2] = { { {c1, 0.f}, {-s1, 0.f} }, { {s1, 0.f}, {c1, 0.f} } };
        C2 RZ[2][2] = { { {c2, -s2}, {0.f, 0.f} }, { {0.f, 0.f}, {c2, s2} } };
        C2 T[2][2], Ul[2][2];
        mm2(T, RY, RX);
        mm2(Ul, RZ, T);
        mm2(U, Ul, U);  // layer applied after accumulated product
    }
    float n00 = fmaf(U[0][0].re, U[0][0].re, U[0][0].im * U[0][0].im);
    float n01 = fmaf(U[0][1].re, U[0][1].re, U[0][1].im * U[0][1].im);
    float n10 = fmaf(U[1][0].re, U[1][0].re, U[1][0].im * U[1][0].im);
    float n11 = fmaf(U[1][1].re, U[1][1].re, U[1][1].im * U[1][1].im);
    float A  = 0.5f * (n00 + n01 - n10 - n11);
    float Bc = 0.5f * (n00 - n01 - n10 + n11);
    float Cs = (fmaf(U[0][0].re, U[0][1].re, U[0][0].im * U[0][1].im))
             - (fmaf(U[1][0].re, U[1][1].re, U[1][0].im * U[1][1].im));
    float a = *pa, b = *pb;
    float k0 = fmaf(a, A, b);
    float kc = a * Bc;
    float ks = a * Cs;

    // -------- depth-4 async-LDS streaming pipeline --------
    __shared__ v4f stage[DEPTH][WPB][32];   // 16 KB: 4 bufs x 8 waves x 512 B

    const int  lane   = threadIdx.x & 31;
    const int  wIn    = threadIdx.x >> 5;
    const long long gwave  = (long long)blockIdx.x * WPB + wIn;
    const long long nwaves = (long long)gridDim.x * WPB;

    const v4f* x4 = (const v4f*)x;
    v4f*       o4 = (v4f*)out;
    const long long N4     = (long long)N >> 2;       // # float4 elements
    const long long ntiles = N4 >> 5;                 // tiles of 32 float4 (one wave-load)

    unsigned offs[DEPTH];                  // addrspace(3) byte offsets of this lane's slots
#pragma unroll
    for (int d = 0; d < DEPTH; ++d)
        offs[d] = (unsigned)(uintptr_t)&stage[d][wIn][lane];

    // prologue: issue up to DEPTH-1 tiles ahead
    int ki = 0;                            // tiles issued
    long long tiss = gwave;                // next tile index to issue
    while (ki < DEPTH - 1 && tiss < ntiles) {
        async_ld_b128(offs[ki & (DEPTH - 1)], x4 + (tiss << 5) + lane);
        ++ki;
        tiss += nwaves;
    }

    int kc_i = 0;                          // tiles consumed
    for (long long t = gwave; t < ntiles; t += nwaves, ++kc_i) {
        if (tiss < ntiles) {
            async_ld_b128(offs[ki & (DEPTH - 1)], x4 + (tiss << 5) + lane);
            ++ki;
            tiss += nwaves;
        }
        wait_async(ki - kc_i - 1);          // tile t has landed in LDS
        v4f v = stage[kc_i & (DEPTH - 1)][wIn][lane];   // ds_load_b128
        v4f r;
        r.x = eval1(v.x, k0, kc, ks);
        r.y = eval1(v.y, k0, kc, ks);
        r.z = eval1(v.z, k0, kc, ks);
        r.w = eval1(v.w, k0, kc, ks);
        __builtin_nontemporal_store(r, o4 + (t << 5) + lane);   // streaming NT store
    }

    // -------- generic tails (empty for N = 16M) --------
    const long long tid      = (long long)blockIdx.x * blockDim.x + threadIdx.x;
    const long long nthreads = (long long)gridDim.x * blockDim.x;
    for (long long i = (ntiles << 5) + tid; i < N4; i += nthreads) {
        v4f v = __builtin_nontemporal_load(x4 + i);
        v4f r;
        r.x = eval1(v.x, k0, kc, ks);
        r.y = eval1(v.y, k0, kc, ks);
        r.z = eval1(v.z, k0, kc, ks);
        r.w = eval1(v.w, k0, kc, ks);
        __builtin_nontemporal_store(r, o4 + i);
    }
    for (long long i = (N4 << 2) + tid; i < (long long)N; i += nthreads) {
        out[i] = eval1(x[i], k0, kc, ks);
    }
}

extern "C" void kernel_launch(void* const* d_in, const int* in_sizes, int n_in,
                              void* d_out, int out_size, void* d_ws, size_t ws_size,
                              hipStream_t stream) {
    const float* x  = (const float*)d_in[0];
    const float* w  = (const float*)d_in[1];
    const float* pa = (const float*)d_in[2];
    const float* pb = (const float*)d_in[3];
    float* out = (float*)d_out;
    const int N = in_sizes[0];
    const int L = in_sizes[1] / 3;

    int blocks = GRID_BLOCKS;
    long long need = ((long long)N + 4LL * BLOCK - 1) / (4LL * BLOCK);
    if (need < blocks) blocks = (int)(need > 0 ? need : 1);

    qpinn_stream_kernel<<<blocks, BLOCK, 0, stream>>>(x, w, L, pa, pb, out, N);
}